// YoloDistLoss_9285719294296
// MI455X (gfx1250) — compile-verified
//
#include <hip/hip_runtime.h>
#include <hip/hip_bf16.h>

#define NB 16
#define NA 5
#define NH 96
#define NW 96
#define NC 80
#define NT 200
#define NCH (6 + NC)              // 86 channels per cell
#define NCELL (NB * NA * NH * NW) // 737280
#define SCALE_F 16.0f
#define IGNORE_THRES_F 0.5f
#define BAD_CONF_W 1.25f
#define PI_F 3.14159265358979323846f

typedef __attribute__((ext_vector_type(2))) float v2f;
typedef __attribute__((ext_vector_type(8))) float v8f;

// ---------------------------------------------------------------------------
// Wave32 full reduction via V_WMMA_F32_16X16X4_F32 (ones-matrix trick).
// A (16x4, f32, 2 VGPRs): lanes 0-15 hold K=0 (vgpr0)/K=1 (vgpr1),
// lanes 16-31 hold K=2/K=3. We put the per-lane value in vgpr0, 0 in vgpr1,
// B = all ones => D[m][n] = v(m) + v(m+16) for every column n.
// Summing the 8 D VGPRs per lane covers rows 0-7 (lanes<16) / 8-15 (lanes>=16);
// one xor-16 shuffle completes the 32-lane sum. Must be called with EXEC all 1s
// (uniform control flow) per the WMMA restrictions.
// ---------------------------------------------------------------------------
__device__ __forceinline__ float wave_sum32(float v) {
#if __has_builtin(__builtin_amdgcn_wmma_f32_16x16x4_f32)
  v2f a;
  a[0] = v;
  a[1] = 0.0f;
  v2f b;
  b[0] = 1.0f;
  b[1] = 1.0f;
  v8f c = {};
  v8f d = __builtin_amdgcn_wmma_f32_16x16x4_f32(
      /*neg_a=*/false, a, /*neg_b=*/false, b,
      /*c_mod=*/(short)0, c, /*reuse_a=*/false, /*reuse_b=*/false);
  float s = 0.0f;
#pragma unroll
  for (int k = 0; k < 8; ++k) s += d[k];
  s += __shfl_xor(s, 16, 32);
  return s;
#else
#pragma unroll
  for (int off = 16; off > 0; off >>= 1) v += __shfl_xor(v, off, 32);
  return v;
#endif
}

__device__ __forceinline__ float inv_tanh_f(float y) {
  if (y <= -1.0f) return -2.0f;
  if (y >= 1.0f) return 2.0f;
  float ys = fminf(fmaxf(y, -1.0f + 1e-6f), 1.0f - 1e-6f);
  return 0.5f * logf((1.0f + ys) / (1.0f - ys));
}

__device__ __forceinline__ float softplus_f(float x) {
  // stable: max(x,0) + log1p(exp(-|x|))
  return fmaxf(x, 0.0f) + log1pf(expf(-fabsf(x)));
}

// ---------------------------------------------------------------------------
// Kernel A: per-target anchor matching + scatter (B*T = 3200 threads)
// ---------------------------------------------------------------------------
__global__ void yolo_scatter(const float* __restrict__ target,
                             const long long* __restrict__ tsizes,
                             const float* __restrict__ anchors,
                             int* __restrict__ mask, int* __restrict__ ignore,
                             int* __restrict__ tcls, float* __restrict__ tx,
                             float* __restrict__ ty, float* __restrict__ tw,
                             float* __restrict__ th, float* __restrict__ tr,
                             int* __restrict__ nGT) {
  int tid = blockIdx.x * blockDim.x + threadIdx.x;
  if (tid >= NB * NT) return;
  int b = tid / NT;
  int t = tid - b * NT;

  const float* trow = target + (size_t)tid * (13 + NC);
  float gx = trow[0] / SCALE_F;
  float gy = trow[1] / SCALE_F;
  float gr = trow[2];
  float gh = trow[3] / SCALE_F;
  float gw = trow[4] / SCALE_F;

  bool valid = (t < (int)tsizes[b]) && (gw != 0.0f) && (gh != 0.0f);

  int gi = min(max((int)gx, 0), NW - 1);
  int gj = min(max((int)gy, 0), NH - 1);

  // rel uses only the first of the 8 points (ref: d = rel[..., :2] - anchor_pts[..., :2])
  float relx = trow[5] / SCALE_F - gx;
  float rely = trow[6] / SCALE_F - gy;

  float dist[NA], sw_a[NA], sh_a[NA], sr_a[NA];
  float best_d = INFINITY;
  int best_n = 0;
#pragma unroll
  for (int a = 0; a < NA; ++a) {
    float sw = anchors[a * 3 + 0] / SCALE_F;
    float sh = anchors[a * 3 + 1] / SCALE_F;
    float sr = anchors[a * 3 + 2];
    sw_a[a] = sw;
    sh_a[a] = sh;
    sr_a[a] = sr;
    float cr = cosf(sr), sn = sinf(sr);
    float dx = relx - (-cr * sw);
    float dy = rely - (-sn * sw);
    float d2 = sqrtf(dx * dx + dy * dy);
    float nrm = ((gh + gw) * 0.5f + (sh + sw) * 0.5f) * 0.5f;
    float q = 4.0f * d2 / nrm;
    dist[a] = q * q;
    if (dist[a] < best_d) { best_d = dist[a]; best_n = a; }  // first-min, like argmin
  }

  if (!valid) return;

#pragma unroll
  for (int a = 0; a < NA; ++a) {
    if (dist[a] < IGNORE_THRES_F) {
      int ic = ((b * NA + a) * NH + gj) * NW + gi;
      atomicOr(&ignore[ic], 1);
    }
  }

  int cidx = ((b * NA + best_n) * NH + gj) * NW + gi;
  atomicOr(&mask[cidx], 1);

  float txv = inv_tanh_f(gx - ((float)gi + 0.5f));
  float tyv = inv_tanh_f(gy - ((float)gj + 0.5f));
  float rd = gr - sr_a[best_n];
  if (rd > PI_F) rd -= 2.0f * PI_F;
  if (rd < -PI_F) rd += 2.0f * PI_F;
  float trv = inv_tanh_f(rd / (PI_F * 0.5f));
  float twv = logf(gw / sw_a[best_n] + 1e-16f);
  float thv = logf(gh / sh_a[best_n] + 1e-16f);

  const float* oh = trow + 13;
  int cbest = 0;
  float cmax = oh[0];
#pragma unroll 4
  for (int j = 1; j < NC; ++j)
    if (oh[j] > cmax) { cmax = oh[j]; cbest = j; }

  // Duplicate-index races mirror JAX's unspecified .set order; acceptable.
  tx[cidx] = txv;
  ty[cidx] = tyv;
  tw[cidx] = twv;
  th[cidx] = thv;
  tr[cidx] = trv;
  tcls[cidx] = cbest;
  atomicAdd(nGT, 1);
}

// ---------------------------------------------------------------------------
// Kernel B: streaming masked reduction over all cells.
// Partials: 0=cnt_mask 1=cnt_false 2=bce_false 3=bce_true 4..8=mse(x,y,w,h,r) 9=nll
// ---------------------------------------------------------------------------
__global__ void __launch_bounds__(256) yolo_reduce(
    const float* __restrict__ pred, const int* __restrict__ mask,
    const int* __restrict__ ignore, const int* __restrict__ tcls,
    const float* __restrict__ tx, const float* __restrict__ ty,
    const float* __restrict__ tw, const float* __restrict__ th,
    const float* __restrict__ tr, float* __restrict__ acc) {
  float p[10];
#pragma unroll
  for (int q = 0; q < 10; ++q) p[q] = 0.0f;

  int stride = gridDim.x * blockDim.x;
  for (int i = blockIdx.x * blockDim.x + threadIdx.x; i < NCELL; i += stride) {
    if (i + stride < NCELL) {
      // global_prefetch_b8 of the next cell's cacheline (speculative is fine)
      __builtin_prefetch(pred + (size_t)(i + stride) * NCH, 0, 0);
    }
    int m = mask[i];
    int ig = ignore[i];
    size_t base = (size_t)i * NCH;
    float conf = pred[base + 0];
    if (m) {
      p[0] += 1.0f;
      p[3] += softplus_f(-conf);
      float px = pred[base + 1], py = pred[base + 2], pr_ = pred[base + 3];
      float ph = pred[base + 4], pw = pred[base + 5];
      float dx = px - tx[i];
      float dy = py - ty[i];
      float dw = pw - tw[i];
      float dh = ph - th[i];
      float dr = pr_ - tr[i];
      p[4] += dx * dx;
      p[5] += dy * dy;
      p[6] += dw * dw;
      p[7] += dh * dh;
      p[8] += dr * dr;
      // log-softmax NLL over 80 classes (rare path, ~0.4% of cells)
      const float* cls = pred + base + 6;
      float mx = cls[0];
#pragma unroll 4
      for (int j = 1; j < NC; ++j) mx = fmaxf(mx, cls[j]);
      float s = 0.0f;
#pragma unroll 4
      for (int j = 0; j < NC; ++j) s += expf(cls[j] - mx);
      int tc = tcls[i];
      p[9] += -(cls[tc] - mx - logf(s));
    } else if (!ig) {
      p[1] += 1.0f;
      p[2] += softplus_f(conf);
    }
  }

  // Uniform control flow from here (EXEC all-ones for WMMA).
  float w[10];
#pragma unroll
  for (int q = 0; q < 10; ++q) w[q] = wave_sum32(p[q]);

  __shared__ float red[8][10];
  int wave = threadIdx.x >> 5;
  int lane = threadIdx.x & 31;
  if (lane == 0) {
#pragma unroll
    for (int q = 0; q < 10; ++q) red[wave][q] = w[q];
  }
  __syncthreads();
  if (threadIdx.x < 10) {
    float s = 0.0f;
#pragma unroll
    for (int wv = 0; wv < 8; ++wv) s += red[wv][threadIdx.x];
    atomicAdd(&acc[threadIdx.x], s);
  }
}

// ---------------------------------------------------------------------------
// Kernel C: finalize scalar loss
// ---------------------------------------------------------------------------
__global__ void yolo_final(const float* __restrict__ acc,
                           const int* __restrict__ nGT,
                           float* __restrict__ out) {
  float n_true = fmaxf(acc[0], 1.0f);
  float n_false = fmaxf(acc[1], 1.0f);
  float bce_false = acc[2] / n_false;
  float bce_true = acc[3] / n_true;
  float mm = (acc[4] + acc[5] + acc[6] + acc[7] + acc[8]) / n_true;
  float lcls = acc[9] / n_true / (float)NB;
  float full = mm + lcls + BAD_CONF_W * bce_false + bce_true;
  out[0] = (*nGT > 0) ? full : (BAD_CONF_W * bce_false);
}

// ---------------------------------------------------------------------------
// Launch: memset(acc+mask+ignore) -> scatter -> reduce -> finalize
// Workspace: [0,64)=acc(16 f32) [64,128)=nGT  then mask, ignore, tcls (int32
// grids) and tx,ty,tw,th,tr (f32 grids), each NCELL elements. ~23.6 MB total.
// ---------------------------------------------------------------------------
extern "C" void kernel_launch(void* const* d_in, const int* in_sizes, int n_in,
                              void* d_out, int out_size, void* d_ws,
                              size_t ws_size, hipStream_t stream) {
  const float* pred = (const float*)d_in[0];
  const float* target = (const float*)d_in[1];
  const long long* tsz = (const long long*)d_in[2];  // int64 in reference
  const float* anchors = (const float*)d_in[3];

  char* ws = (char*)d_ws;
  float* acc = (float*)ws;          // 16 floats
  int* nGT = (int*)(ws + 64);       // 1 int (padded)
  int* mask = (int*)(ws + 128);
  int* ignore = mask + NCELL;
  int* tcls = ignore + NCELL;
  float* tx = (float*)(tcls + NCELL);
  float* ty = tx + NCELL;
  float* tw = ty + NCELL;
  float* th = tw + NCELL;
  float* tr = th + NCELL;

  // Zero accumulators + nGT + mask + ignore. t-grids are only read where mask
  // is set, and every mask-setting thread also writes them, so no init needed.
  size_t zero_bytes = 128 + 2 * (size_t)NCELL * sizeof(int);
  hipMemsetAsync(d_ws, 0, zero_bytes, stream);

  yolo_scatter<<<(NB * NT + 255) / 256, 256, 0, stream>>>(
      target, tsz, anchors, mask, ignore, tcls, tx, ty, tw, th, tr, nGT);

  yolo_reduce<<<NCELL / 256, 256, 0, stream>>>(pred, mask, ignore, tcls, tx,
                                               ty, tw, th, tr, acc);

  yolo_final<<<1, 1, 0, stream>>>(acc, nGT, (float*)d_out);
}